// Net_8564164788766
// MI455X (gfx1250) — compile-verified
//
#include <hip/hip_runtime.h>
#include <hip/hip_bf16.h>

// ---------------------------------------------------------------------------
// GCN pipeline for MI455X (gfx1250, wave32).
//  - Edge aggregation is the dominant cost; its working set (xw/agg, 12.8 MB
//    each) is L2-resident (192 MB L2), so we scatter with f32 global atomics.
//  - Dense node transforms use native fp32 WMMA: V_WMMA_F32_16X16X4_F32.
// ---------------------------------------------------------------------------

typedef float v2f __attribute__((ext_vector_type(2)));
typedef float v8f __attribute__((ext_vector_type(8)));

#define N_NODES  50000
#define N_EDGES  1600000
#define N_GRAPHS 512
#define WIDTH    64
#define LAYERS   4

__device__ __forceinline__ float elu_f(float x) {
    return x > 0.0f ? x : (__expf(x) - 1.0f);
}

// monotone float -> uint mapping so unsigned atomicMax == float max
__device__ __forceinline__ unsigned fmap(float f) {
    unsigned u = __float_as_uint(f);
    return (u & 0x80000000u) ? ~u : (u | 0x80000000u);
}
__device__ __forceinline__ float funmap(unsigned u) {
    return (u & 0x80000000u) ? __uint_as_float(u & 0x7FFFFFFFu)
                             : __uint_as_float(~u);
}

// ---------------------------------------------------------------------------
__global__ void k_zero(float* __restrict__ p, int n) {
    int i = blockIdx.x * blockDim.x + threadIdx.x;
    if (i < n) p[i] = 0.0f;
}

__global__ void k_deg(const int* __restrict__ dst, float* __restrict__ deg, int E) {
    int e = blockIdx.x * blockDim.x + threadIdx.x;
    if (e < E) atomicAdd(deg + dst[e], 1.0f);
}

__global__ void k_dinv(const float* __restrict__ deg, float* __restrict__ dinv, int n) {
    int i = blockIdx.x * blockDim.x + threadIdx.x;
    if (i < n) dinv[i] = rsqrtf(deg[i] + 1.0f);   // +1 self-loop
}

__global__ void k_coef(const int* __restrict__ src, const int* __restrict__ dst,
                       const float* __restrict__ dinv, float* __restrict__ coef, int E) {
    int e = blockIdx.x * blockDim.x + threadIdx.x;
    if (e < E) coef[e] = dinv[src[e]] * dinv[dst[e]];
}

// Layer 1 transform: x[N,2] @ W1[2,32]  (K=2 -> plain FMA)
__global__ void k_lin2(const float* __restrict__ x, const float* __restrict__ W1,
                       float* __restrict__ xw, int n) {
    int gid = blockIdx.x * blockDim.x + threadIdx.x;
    if (gid >= n * 32) return;
    int node = gid >> 5, j = gid & 31;
    xw[gid] = x[node * 2 + 0] * W1[j] + x[node * 2 + 1] * W1[32 + j];
}

// ---------------------------------------------------------------------------
// WMMA fp32 GEMM:  XW[nrows,64] = H[nrows,K] @ W[K,64]
// block = 128 threads = 4 waves; wave w owns a 16x16 column tile.
// A 16x4 f32 fragment: lane(0..15)=M, half selects K pair {0,1}/{2,3}.
// B 4x16 f32 fragment: lane(0..15)=N, same half/K mapping.
// D: vgpr r, lane half -> row M = r + 8*half, col N = lane&15.
// ---------------------------------------------------------------------------
template <int K>
__global__ __launch_bounds__(128) void k_gemm_wmma(const float* __restrict__ H,
                                                   const float* __restrict__ W,
                                                   float* __restrict__ XW) {
    const int lane = threadIdx.x & 31;
    const int wv   = threadIdx.x >> 5;        // 0..3 -> column tile
    const int half = lane >> 4;               // 0/1
    const int idx  = lane & 15;               // m (for A) or n (for B/D)
    const int row0 = blockIdx.x * 16;
    const int c0   = wv * 16;

    v8f acc = {};
    const float* hrow = H + (size_t)(row0 + idx) * K;
#pragma unroll
    for (int kk = 0; kk < K; kk += 4) {
        const int ka = kk + 2 * half;
        v2f a, b;
        a.x = hrow[ka];
        a.y = hrow[ka + 1];
        b.x = W[(size_t)ka * 64 + c0 + idx];
        b.y = W[(size_t)(ka + 1) * 64 + c0 + idx];
        acc = __builtin_amdgcn_wmma_f32_16x16x4_f32(
            /*neg_a=*/false, a, /*neg_b=*/false, b,
            /*c_mod=*/(short)0, acc, /*reuse_a=*/false, /*reuse_b=*/false);
    }
#pragma unroll
    for (int r = 0; r < 8; ++r)
        XW[(size_t)(row0 + r + 8 * half) * 64 + c0 + idx] = acc[r];
}

// ---------------------------------------------------------------------------
// Edge scatter-add:  agg[dst] += xw[src] * coef[e]
// W/4 threads per edge, float4 per thread (coalesced gather, f32 atomics).
// ---------------------------------------------------------------------------
template <int W>
__global__ void k_agg(const float* __restrict__ xw, const int* __restrict__ src,
                      const int* __restrict__ dst, const float* __restrict__ coef,
                      float* __restrict__ agg, int E) {
    const int TPE = W / 4;
    int gid = blockIdx.x * blockDim.x + threadIdx.x;
    int e = gid / TPE;
    if (e >= E) return;
    int f = (gid % TPE) * 4;
    float c = coef[e];
    const float4 v = *reinterpret_cast<const float4*>(xw + (size_t)src[e] * W + f);
    float* a = agg + (size_t)dst[e] * W + f;
    atomicAdd(a + 0, v.x * c);
    atomicAdd(a + 1, v.y * c);
    atomicAdd(a + 2, v.z * c);
    atomicAdd(a + 3, v.w * c);
}

// h[n,j] = elu( agg[n,j] + xw[n,j]*dinv[n]^2 + b[j] )
template <int W>
__global__ void k_finalize(const float* __restrict__ agg, const float* __restrict__ xw,
                           const float* __restrict__ dinv, const float* __restrict__ b,
                           float* __restrict__ h, int n) {
    int gid = blockIdx.x * blockDim.x + threadIdx.x;
    if (gid >= n * W) return;
    int node = gid / W, j = gid % W;
    float di = dinv[node];
    h[gid] = elu_f(agg[gid] + xw[gid] * di * di + b[j]);
}

// segment_max pooling via monotone-uint atomicMax
__global__ void k_pool(const float* __restrict__ h, const int* __restrict__ batch,
                       unsigned* __restrict__ g, int n) {
    int gid = blockIdx.x * blockDim.x + threadIdx.x;
    if (gid >= n * WIDTH) return;
    int node = gid >> 6, j = gid & 63;
    atomicMax(g + (size_t)batch[node] * WIDTH + j, fmap(h[gid]));
}

// FC head: one workgroup (64 threads) per graph; log_softmax output.
__global__ __launch_bounds__(64) void k_head(const unsigned* __restrict__ gpool,
                                             const float* __restrict__ fcW1, const float* __restrict__ fcb1,
                                             const float* __restrict__ fcW2, const float* __restrict__ fcb2,
                                             const float* __restrict__ fcW3, const float* __restrict__ fcb3,
                                             float* __restrict__ out) {
    __shared__ float g[64];
    __shared__ float t1[64];
    __shared__ float t2[32];
    const int gi = blockIdx.x, j = threadIdx.x;
    g[j] = funmap(gpool[gi * 64 + j]);
    __syncthreads();
    float s = fcb1[j];
#pragma unroll 8
    for (int k = 0; k < 64; ++k) s += g[k] * fcW1[k * 64 + j];
    t1[j] = elu_f(s);
    __syncthreads();
    if (j < 32) {
        float s2 = fcb2[j];
#pragma unroll 8
        for (int k = 0; k < 64; ++k) s2 += t1[k] * fcW2[k * 32 + j];
        t2[j] = elu_f(s2);
    }
    __syncthreads();
    if (j < 2) {
        float s3 = fcb3[j];
#pragma unroll 8
        for (int k = 0; k < 32; ++k) s3 += t2[k] * fcW3[k * 2 + j];
        t1[j] = s3;   // reuse as logit store
    }
    __syncthreads();
    if (j == 0) {
        float a = t1[0], b = t1[1];
        float m = fmaxf(a, b);
        float lse = m + __logf(__expf(a - m) + __expf(b - m));
        out[gi * 2 + 0] = a - lse;
        out[gi * 2 + 1] = b - lse;
    }
}

// ---------------------------------------------------------------------------
static inline char* carve(char*& p, size_t bytes) {
    char* r = p;
    p += (bytes + 255) & ~(size_t)255;
    return r;
}

extern "C" void kernel_launch(void* const* d_in, const int* in_sizes, int n_in,
                              void* d_out, int out_size, void* d_ws, size_t ws_size,
                              hipStream_t stream) {
    (void)n_in; (void)out_size; (void)ws_size;
    // inputs (setup_inputs order)
    const float* x      = (const float*)d_in[0];
    const int*   ei     = (const int*)d_in[1];
    const int*   batch  = (const int*)d_in[2];
    const float* W1     = (const float*)d_in[3];
    const float* b1     = (const float*)d_in[4];
    const float* W2     = (const float*)d_in[5];
    const float* b2     = (const float*)d_in[6];
    const float* convW  = (const float*)d_in[7];
    const float* convB  = (const float*)d_in[8];
    const float* fcW1   = (const float*)d_in[9];
    const float* fcb1   = (const float*)d_in[10];
    const float* fcW2   = (const float*)d_in[11];
    const float* fcb2   = (const float*)d_in[12];
    const float* fcW3   = (const float*)d_in[13];
    const float* fcb3   = (const float*)d_in[14];
    float* out = (float*)d_out;

    const int N = in_sizes[2];           // 50000
    const int E = in_sizes[1] / 2;       // 1600000
    const int* src = ei;
    const int* dst = ei + E;

    // workspace carving (~45 MB)
    char* p = (char*)d_ws;
    float*    deg   = (float*)carve(p, (size_t)N * 4);
    float*    dinv  = (float*)carve(p, (size_t)N * 4);
    float*    coef  = (float*)carve(p, (size_t)E * 4);
    float*    xw    = (float*)carve(p, (size_t)N * WIDTH * 4);
    float*    agg   = (float*)carve(p, (size_t)N * WIDTH * 4);
    float*    h     = (float*)carve(p, (size_t)N * WIDTH * 4);
    unsigned* gpool = (unsigned*)carve(p, (size_t)N_GRAPHS * WIDTH * 4);

    const int B = 256;
    auto blocks = [](int n, int b) { return (n + b - 1) / b; };

    // degrees + normalization
    k_zero<<<blocks(N, B), B, 0, stream>>>(deg, N);
    k_deg<<<blocks(E, B), B, 0, stream>>>(dst, deg, E);
    k_dinv<<<blocks(N, B), B, 0, stream>>>(deg, dinv, N);
    k_coef<<<blocks(E, B), B, 0, stream>>>(src, dst, dinv, coef, E);

    // ---- layer 1: [N,2] -> [N,32] ----
    k_lin2<<<blocks(N * 32, B), B, 0, stream>>>(x, W1, xw, N);
    k_zero<<<blocks(N * 32, B), B, 0, stream>>>(agg, N * 32);
    k_agg<32><<<blocks(E * 8, B), B, 0, stream>>>(xw, src, dst, coef, agg, E);
    k_finalize<32><<<blocks(N * 32, B), B, 0, stream>>>(agg, xw, dinv, b1, h, N);

    // ---- layer 2: [N,32] -> [N,64] (WMMA, K=32) ----
    k_gemm_wmma<32><<<N / 16, 128, 0, stream>>>(h, W2, xw);
    k_zero<<<blocks(N * 64, B), B, 0, stream>>>(agg, N * 64);
    k_agg<64><<<blocks(E * 16, B), B, 0, stream>>>(xw, src, dst, coef, agg, E);
    k_finalize<64><<<blocks(N * 64, B), B, 0, stream>>>(agg, xw, dinv, b2, h, N);

    // ---- conv layers: [N,64] -> [N,64] (WMMA, K=64) ----
    for (int l = 0; l < LAYERS; ++l) {
        k_gemm_wmma<64><<<N / 16, 128, 0, stream>>>(h, convW + (size_t)l * 64 * 64, xw);
        k_zero<<<blocks(N * 64, B), B, 0, stream>>>(agg, N * 64);
        k_agg<64><<<blocks(E * 16, B), B, 0, stream>>>(xw, src, dst, coef, agg, E);
        k_finalize<64><<<blocks(N * 64, B), B, 0, stream>>>(agg, xw, dinv, convB + l * 64, h, N);
    }

    // ---- pooling + head ----
    k_zero<<<blocks(N_GRAPHS * WIDTH, B), B, 0, stream>>>((float*)gpool, N_GRAPHS * WIDTH);
    k_pool<<<blocks(N * WIDTH, B), B, 0, stream>>>(h, batch, gpool, N);
    k_head<<<N_GRAPHS, 64, 0, stream>>>(gpool, fcW1, fcb1, fcW2, fcb2, fcW3, fcb3, out);
}